// HardGate_57466662420621
// MI455X (gfx1250) — compile-verified
//
#include <hip/hip_runtime.h>
#include <hip/hip_bf16.h>
#include <stdint.h>

// Problem constants (from reference): 16384 tokens x 4096 fp32 features, 16 experts.
#define N_TOK   16384
#define DIM     4096
#define NEXP    16
#define CHUNK   256            // tokens per histogram/rank block
#define NBLK    (N_TOK / CHUNK) // 64

// ---------------------------------------------------------------------------
// Kernel 1: per-chunk expert histogram.
// ---------------------------------------------------------------------------
__global__ __launch_bounds__(CHUNK) void hg_hist(const int* __restrict__ mapping,
                                                 int* __restrict__ blockHist) {
    __shared__ int h[NEXP];
    const int t = threadIdx.x;
    if (t < NEXP) h[t] = 0;
    __syncthreads();
    const int i = blockIdx.x * CHUNK + t;
    const int e = mapping[i];
    atomicAdd(&h[e], 1);
    __syncthreads();
    if (t < NEXP) blockHist[blockIdx.x * NEXP + t] = h[t];
}

// ---------------------------------------------------------------------------
// Kernel 2: single tiny block. Per-expert totals -> expert_splits output,
// exclusive expert offsets, and stable per-block bases:
//   blockBase[b][e] = off[e] + sum_{b'<b} blockHist[b'][e]
// ---------------------------------------------------------------------------
__global__ __launch_bounds__(32) void hg_scan(const int* __restrict__ blockHist,
                                              int* __restrict__ blockBase,
                                              int* __restrict__ splits_out) {
    __shared__ int tot[NEXP];
    __shared__ int off[NEXP];
    const int e = threadIdx.x;
    if (e < NEXP) {
        int run = 0;
        for (int b = 0; b < NBLK; ++b) {
            blockBase[b * NEXP + e] = run;
            run += blockHist[b * NEXP + e];
        }
        tot[e] = run;
    }
    __syncthreads();
    if (e == 0) {
        int run = 0;
        for (int k = 0; k < NEXP; ++k) { off[k] = run; run += tot[k]; }
    }
    __syncthreads();
    if (e < NEXP) {
        splits_out[e] = tot[e];
        const int o = off[e];
        for (int b = 0; b < NBLK; ++b) blockBase[b * NEXP + e] += o;
    }
}

// ---------------------------------------------------------------------------
// Kernel 3: stable in-chunk rank -> destination slot.
//   dst = blockBase[b][e] + #{t' < t in chunk : map[t'] == e}
// Writes reverse_sorted_sample_pos[i] = dst and gather index src_for[dst] = i.
// ---------------------------------------------------------------------------
__global__ __launch_bounds__(CHUNK) void hg_rank(const int* __restrict__ mapping,
                                                 const int* __restrict__ blockBase,
                                                 int* __restrict__ reverse_out,
                                                 int* __restrict__ src_for) {
    __shared__ int m[CHUNK];
    const int t = threadIdx.x;
    const int i = blockIdx.x * CHUNK + t;
    const int e = mapping[i];
    m[t] = e;
    __syncthreads();
    int rank = 0;
    for (int k = 0; k < t; ++k) rank += (m[k] == e) ? 1 : 0;
    const int dst = blockBase[blockIdx.x * NEXP + e] + rank;
    reverse_out[i] = dst;
    src_for[dst]   = i;
}

// ---------------------------------------------------------------------------
// Kernel 4: the 512 MiB row gather. One block per output row (16 KB).
// Stream the row through LDS with CDNA5 async global<->LDS b128 DMA:
// each lane issues 4x 16B async loads into LDS, waits ASYNCcnt==0, then
// 4x 16B async stores from the same LDS bytes. No VGPR data registers used
// for the payload. Per-lane self-dependency only -> no workgroup barrier.
// IOFFSET is added to BOTH the LDS and global address (CDNA5 ISA
// pseudocode), so one base address pair serves all 4 sweeps.
// ---------------------------------------------------------------------------
__global__ __launch_bounds__(256) void hg_copy(const float* __restrict__ in,
                                               const int* __restrict__ src_for,
                                               float* __restrict__ out) {
    __shared__ float buf[DIM]; // 16 KB staging
    const int d   = blockIdx.x;
    const int src = src_for[d];
    const unsigned t = threadIdx.x;

    // Flat shared-aperture address: low 32 bits are the LDS byte offset.
    const unsigned lds_off =
        (unsigned)(uintptr_t)(&buf[0]) + t * 16u;
    const unsigned long long gsrc =
        (unsigned long long)(uintptr_t)(in  + (size_t)src * DIM) + (unsigned long long)t * 16ull;
    const unsigned long long gdst =
        (unsigned long long)(uintptr_t)(out + (size_t)d   * DIM) + (unsigned long long)t * 16ull;

    asm volatile(
        "global_load_async_to_lds_b128 %0, %1, off\n\t"
        "global_load_async_to_lds_b128 %0, %1, off offset:4096\n\t"
        "global_load_async_to_lds_b128 %0, %1, off offset:8192\n\t"
        "global_load_async_to_lds_b128 %0, %1, off offset:12288\n\t"
        "s_wait_asynccnt 0x0\n\t"
        "global_store_async_from_lds_b128 %2, %0, off\n\t"
        "global_store_async_from_lds_b128 %2, %0, off offset:4096\n\t"
        "global_store_async_from_lds_b128 %2, %0, off offset:8192\n\t"
        "global_store_async_from_lds_b128 %2, %0, off offset:12288\n\t"
        "s_wait_asynccnt 0x0"
        :
        : "v"(lds_off), "v"(gsrc), "v"(gdst)
        : "memory");
}

// ---------------------------------------------------------------------------
// Host-side launcher. d_in[0]=inputs (fp32, 16384*4096), d_in[1]=mapping (i32).
// d_out (float*) layout = dispatched | expert_splits (int bits) | reverse (int bits).
// Workspace: blockHist (4 KB) | blockBase (4 KB) | src_for (64 KB) = 72 KB.
// ---------------------------------------------------------------------------
extern "C" void kernel_launch(void* const* d_in, const int* in_sizes, int n_in,
                              void* d_out, int out_size, void* d_ws, size_t ws_size,
                              hipStream_t stream) {
    const float* inputs  = (const float*)d_in[0];
    const int*   mapping = (const int*)d_in[1];

    float* out_f      = (float*)d_out;
    float* dispatched = out_f;                                   // 16384*4096 fp32
    int*   splits_out = (int*)(out_f + (size_t)N_TOK * DIM);     // 16 i32 (bit-cast slot)
    int*   reverse_out = splits_out + NEXP;                      // 16384 i32

    char* ws = (char*)d_ws;
    int* blockHist = (int*)(ws);                   // NBLK*NEXP ints
    int* blockBase = (int*)(ws + 4096);            // NBLK*NEXP ints
    int* src_for   = (int*)(ws + 8192);            // N_TOK ints

    hg_hist<<<dim3(NBLK), dim3(CHUNK), 0, stream>>>(mapping, blockHist);
    hg_scan<<<dim3(1), dim3(32), 0, stream>>>(blockHist, blockBase, splits_out);
    hg_rank<<<dim3(NBLK), dim3(CHUNK), 0, stream>>>(mapping, blockBase, reverse_out, src_for);
    hg_copy<<<dim3(N_TOK), dim3(256), 0, stream>>>(inputs, src_for, dispatched);
}